// Sinkhorn_9337258902036
// MI455X (gfx1250) — compile-verified
//
#include <hip/hip_runtime.h>
#include <hip/hip_bf16.h>

#define EPS 1e-4f
#define BN 64
#define NN 1024
#define MM 1024

typedef __attribute__((ext_vector_type(2))) float v2f;
typedef __attribute__((ext_vector_type(8))) float v8f;

// a[b][n] = (n < nrows[b]) ? 1 : 0   (row mask folded into the row factor)
__global__ void sinkhorn_init_a(float* __restrict__ avec,
                                const int* __restrict__ nrows) {
    int idx = blockIdx.x * blockDim.x + threadIdx.x;   // [0, B*N)
    int b = idx >> 10;          // / NN
    int n = idx & (NN - 1);
    avec[idx] = (n < nrows[b]) ? 1.0f : 0.0f;
}

// Even iteration: b[m] = 1 / sum_n (s+EPS)[b][n][m] * a[b][n], zero for m>=ncols.
// One wave handles 16 columns; the dot over N runs as 256 WMMA f32 16x16x4 steps:
//   A(16x4) = a-chunk replicated down rows, B(4x16) = s-tile, D accumulates.
__global__ void sinkhorn_colsum_wmma(const float* __restrict__ s,
                                     const float* __restrict__ avec,
                                     float* __restrict__ bvec,
                                     const int* __restrict__ ncols) {
    int wave = blockIdx.x * (blockDim.x >> 5) + (threadIdx.x >> 5);
    int lane = threadIdx.x & 31;
    int wavesPerBatch = MM / 16;              // 64
    int b  = wave / wavesPerBatch;            // wave-uniform
    int m0 = (wave % wavesPerBatch) * 16;

    const float* sb = s    + (size_t)b * NN * MM;
    const float* ab = avec + (size_t)b * NN;

    // A-matrix 16x4 f32 layout: VGPR0 = K0(lanes0-15)/K2(lanes16-31), VGPR1 = K1/K3.
    // B-matrix 4x16 f32 layout: VGPR0 = row K0(lanes0-15)/K2(lanes16-31), VGPR1 = K1/K3.
    int half = (lane >= 16) ? 2 : 0;
    int col  = m0 + (lane & 15);

    v8f acc = {};
    for (int n0 = 0; n0 < NN; n0 += 4) {
        // prefetch a couple of chunks ahead (global_prefetch_b8)
        __builtin_prefetch(&sb[(size_t)(n0 + 8 + half) * MM + col], 0, 0);

        v2f A, Bm;
        A[0]  = ab[n0 + half];
        A[1]  = ab[n0 + 1 + half];
        Bm[0] = sb[(size_t)(n0 + half)     * MM + col] + EPS;
        Bm[1] = sb[(size_t)(n0 + 1 + half) * MM + col] + EPS;
        // D(16x16) += A(16x4) x B(4x16); every D row holds the same 16 column sums
        acc = __builtin_amdgcn_wmma_f32_16x16x4_f32(
            /*neg_a=*/false, A, /*neg_b=*/false, Bm,
            /*c_mod=*/(short)0, acc, /*reuse_a=*/false, /*reuse_b=*/false);
    }

    if (lane < 16) {
        int m = m0 + lane;
        float denom = acc[0];   // D row 0: VGPR0, lanes 0-15 = columns m0..m0+15
        bvec[(size_t)b * MM + m] = (m < ncols[b]) ? (1.0f / denom) : 0.0f;
    }
}

// Odd iteration: a[n] = 1 / sum_m (s+EPS)[b][n][m] * b[b][m], zero for n>=nrows.
// One wave per row; float4 loads + wave32 xor-shuffle reduction.
__global__ void sinkhorn_rowsum(const float* __restrict__ s,
                                const float* __restrict__ bvec,
                                float* __restrict__ avec,
                                const int* __restrict__ nrows) {
    int wave = blockIdx.x * (blockDim.x >> 5) + (threadIdx.x >> 5);
    int lane = threadIdx.x & 31;
    int b = wave >> 10;          // / NN
    int n = wave & (NN - 1);

    const float4* srow4 = (const float4*)(s    + ((size_t)b * NN + n) * MM);
    const float4* bb4   = (const float4*)(bvec + (size_t)b * MM);

    float sum = 0.0f;
    for (int q = lane; q < MM / 4; q += 32) {
        float4 sv = srow4[q];
        float4 bv = bb4[q];
        sum = fmaf(sv.x + EPS, bv.x, sum);
        sum = fmaf(sv.y + EPS, bv.y, sum);
        sum = fmaf(sv.z + EPS, bv.z, sum);
        sum = fmaf(sv.w + EPS, bv.w, sum);
    }
    for (int off = 16; off; off >>= 1)
        sum += __shfl_xor(sum, off, 32);

    if (lane == 0)
        avec[(size_t)b * NN + n] = (n < nrows[b]) ? (1.0f / sum) : 0.0f;
}

// out = (s+EPS) * a[n] * b[m]  (zeroed factors reproduce the masked zeros)
__global__ void sinkhorn_finalize(const float* __restrict__ s,
                                  const float* __restrict__ avec,
                                  const float* __restrict__ bvec,
                                  float* __restrict__ out) {
    size_t idx = (size_t)blockIdx.x * blockDim.x + threadIdx.x;  // quad index
    size_t q_per_row = MM / 4;
    size_t rown = idx / q_per_row;           // [0, B*N)
    size_t m4   = idx % q_per_row;
    size_t b    = rown >> 10;

    float a = avec[rown];
    float4 bv = ((const float4*)(bvec + b * MM))[m4];
    float4 sv = ((const float4*)s)[idx];

    float4 o;
    o.x = (sv.x + EPS) * a * bv.x;
    o.y = (sv.y + EPS) * a * bv.y;
    o.z = (sv.z + EPS) * a * bv.z;
    o.w = (sv.w + EPS) * a * bv.w;
    ((float4*)out)[idx] = o;
}

extern "C" void kernel_launch(void* const* d_in, const int* in_sizes, int n_in,
                              void* d_out, int out_size, void* d_ws, size_t ws_size,
                              hipStream_t stream) {
    const float* s     = (const float*)d_in[0];
    const int*   nrows = (const int*)d_in[1];
    const int*   ncols = (const int*)d_in[2];
    float*       out   = (float*)d_out;

    float* avec = (float*)d_ws;              // [B][N] = 256 KB
    float* bvec = avec + (size_t)BN * NN;    // [B][M] = 256 KB

    // row mask -> a
    sinkhorn_init_a<<<(BN * NN) / 256, 256, 0, stream>>>(avec, nrows);

    // 5 x (even column-normalize, odd row-normalize) as factored vector updates
    const int colsumBlocks = (BN * (MM / 16) * 32) / 256;   // 4096 waves -> 512 blocks
    const int rowsumBlocks = (BN * NN * 32) / 256;          // 65536 waves -> 8192 blocks
    for (int it = 0; it < 5; ++it) {
        sinkhorn_colsum_wmma<<<colsumBlocks, 256, 0, stream>>>(s, avec, bvec, ncols);
        sinkhorn_rowsum<<<rowsumBlocks, 256, 0, stream>>>(s, bvec, avec, nrows);
    }

    // single fused write-out pass
    const size_t quads = (size_t)BN * NN * MM / 4;          // 16M
    sinkhorn_finalize<<<(unsigned)(quads / 256), 256, 0, stream>>>(s, avec, bvec, out);
}